// SentenceEmbedding_38568806318791
// MI455X (gfx1250) — compile-verified
//
#include <hip/hip_runtime.h>
#include <hip/hip_bf16.h>

typedef float v2f __attribute__((ext_vector_type(2)));
typedef float v8f __attribute__((ext_vector_type(8)));

#define SEQ_T 65536

__device__ __forceinline__ float sigmoid_fast(float x) {
    x = fminf(fmaxf(x, -20.0f), 20.0f);
    return 1.0f / (1.0f + __expf(-x));
}
__device__ __forceinline__ float tanh_fast(float x) {
    x = fminf(fmaxf(x, -20.0f), 20.0f);
    float e = __expf(-2.0f * x);
    return (1.0f - e) / (1.0f + e);
}

// ---------------------------------------------------------------------------
// C[T,N] = A[T,K] @ W[N,K]^T + bias[N]   via V_WMMA_F32_16X16X4_F32
// One wave per 16x16 output tile; 4 M-tiles per 128-thread block.
// A frag (16x4):  lane<16 -> K+0..1 of row (lane&15); lane>=16 -> K+2..3
// B frag (4x16):  B[k][n] = W[n0+n][k]; same half-wave K split, n = lane&15
// C frag (16x16): vgpr r -> M = r + (lane<16 ? 0 : 8), N = lane&15
// ---------------------------------------------------------------------------
__global__ void xg_gemm_wmma(const float* __restrict__ A,
                             const float* __restrict__ W,
                             const float* __restrict__ bias,
                             float* __restrict__ C,
                             int K, int N) {
    const int lane = threadIdx.x & 31;
    const int wave = threadIdx.x >> 5;
    const int m0 = (blockIdx.x * 4 + wave) * 16;
    const int n0 = blockIdx.y * 16;
    const int r15 = lane & 15;
    const int koff = (lane < 16) ? 0 : 2;

    const float* aRow = A + (size_t)(m0 + r15) * K + koff;
    const float* bRow = W + (size_t)(n0 + r15) * K + koff;

    v8f acc = {};
    for (int k = 0; k < K; k += 4) {
        v2f a, b;
        a.x = aRow[k];
        a.y = aRow[k + 1];
        b.x = bRow[k];
        b.y = bRow[k + 1];
        // (neg_a, A, neg_b, B, c_mod, C, reuse_a, reuse_b)
        acc = __builtin_amdgcn_wmma_f32_16x16x4_f32(false, a, false, b,
                                                    (short)0, acc, false, false);
    }

    const float bn = bias[n0 + r15];
    const int rbase = (lane < 16) ? 0 : 8;
    float* cPtr = C + (size_t)(m0 + rbase) * N + (n0 + r15);
#pragma unroll
    for (int r = 0; r < 8; ++r) {
        cPtr[(size_t)r * N] = acc[r] + bn;
    }
}

// ---------------------------------------------------------------------------
// GRU scan, H=16. One wave; zero barriers. Lane j and lane j+16 cooperate on
// gate row j: each accumulates 8 of the 16 K-terms, combined via shfl_xor(16).
// Whh rows live in VGPRs; h lives in lanes 0..15 (mirrored in 16..31) and is
// broadcast with __shfl. xg layout [T,48] (r,z,n stacked).
// ---------------------------------------------------------------------------
__global__ void gru_scan16(const float* __restrict__ xg,
                           const float* __restrict__ Whh,  // [48,16]
                           const float* __restrict__ bhh,  // [48]
                           const float* __restrict__ h0,   // [16] or null -> ones
                           float* __restrict__ y,          // [T,16]
                           float* __restrict__ hT) {       // [16] or null
    const int lane = threadIdx.x;
    const int j = lane & 15;
    const int koff = (lane < 16) ? 0 : 8;

    float wr[8], wz[8], wn[8];
#pragma unroll
    for (int k = 0; k < 8; ++k) {
        wr[k] = Whh[(j) * 16 + koff + k];
        wz[k] = Whh[(16 + j) * 16 + koff + k];
        wn[k] = Whh[(32 + j) * 16 + koff + k];
    }
    const float br = bhh[j], bz = bhh[16 + j], bn = bhh[32 + j];

    float h = h0 ? h0[j] : 1.0f;

    for (int t = 0; t < SEQ_T; ++t) {
        const float* xrow = xg + (size_t)t * 48;
        float xr = xrow[j];
        float xz = xrow[16 + j];
        float xn = xrow[32 + j];
        int tp = t + 48 < SEQ_T ? t + 48 : SEQ_T - 1;
        __builtin_prefetch(xg + (size_t)tp * 48 + j, 0, 1);

        float ar = 0.0f, az = 0.0f, an = 0.0f;
#pragma unroll
        for (int k = 0; k < 8; ++k) {
            float hk = __shfl(h, koff + k, 32);
            ar = fmaf(wr[k], hk, ar);
            az = fmaf(wz[k], hk, az);
            an = fmaf(wn[k], hk, an);
        }
        // combine the two half-wave partial sums
        ar += __shfl_xor(ar, 16, 32);
        az += __shfl_xor(az, 16, 32);
        an += __shfl_xor(an, 16, 32);

        float r = sigmoid_fast(xr + ar + br);
        float z = sigmoid_fast(xz + az + bz);
        float n = tanh_fast(xn + r * (an + bn));
        h = (1.0f - z) * n + z * h;

        if (lane < 16) y[(size_t)t * 16 + j] = h;
    }
    if (hT && lane < 16) hT[j] = h;
}

// ---------------------------------------------------------------------------
// GRU scan, H=64. 192 threads: thread g owns gate row g (Whh row in VGPRs).
// h kept in LDS (broadcast reads, conflict-free); 2 barriers per step.
// ---------------------------------------------------------------------------
__global__ void gru_scan64(const float* __restrict__ xg,   // [T,192]
                           const float* __restrict__ Whh,  // [192,64]
                           const float* __restrict__ bhh,  // [192]
                           float* __restrict__ outH) {     // [64]
    const int g = threadIdx.x;  // 0..191
    __shared__ float h_s[64];
    __shared__ float act[192];
    __shared__ float xn_s[64];

    float w[64];
#pragma unroll
    for (int k = 0; k < 64; ++k) w[k] = Whh[(size_t)g * 64 + k];
    const float b = bhh[g];

    if (g < 64) h_s[g] = 1.0f;
    __syncthreads();

    for (int t = 0; t < SEQ_T; ++t) {
        float xv = xg[(size_t)t * 192 + g];
        int tp = t + 16 < SEQ_T ? t + 16 : SEQ_T - 1;
        __builtin_prefetch(xg + (size_t)tp * 192 + g, 0, 1);

        float a = 0.0f;
#pragma unroll
        for (int k = 0; k < 64; ++k) a = fmaf(w[k], h_s[k], a);
        a += b;

        if (g < 128) {
            act[g] = sigmoid_fast(xv + a);  // r (g<64) or z (64<=g<128)
        } else {
            act[g] = a;        // hn (recurrent part incl. bhh_n)
            xn_s[g - 128] = xv;
        }
        __syncthreads();  // h_s reads done; act/xn visible

        if (g < 64) {
            float r = act[g];
            float z = act[64 + g];
            float n = tanh_fast(xn_s[g] + r * act[128 + g]);
            h_s[g] = (1.0f - z) * n + z * h_s[g];
        }
        __syncthreads();  // new h visible for next step
    }
    if (g < 64) outH[g] = h_s[g];
}

// ---------------------------------------------------------------------------
extern "C" void kernel_launch(void* const* d_in, const int* in_sizes, int n_in,
                              void* d_out, int out_size, void* d_ws, size_t ws_size,
                              hipStream_t stream) {
    (void)in_sizes; (void)n_in; (void)out_size; (void)ws_size;
    const float* input = (const float*)d_in[0];
    // per-layer params: [1..4]=c1l0, [5..8]=c1l1, [9..12]=d1l0, [13..16]=d1l1, [17..20]=d2
    const float* Wih[5]; const float* Whh[5]; const float* bih[5]; const float* bhh[5];
    for (int l = 0; l < 5; ++l) {
        Wih[l] = (const float*)d_in[1 + 4 * l + 0];
        Whh[l] = (const float*)d_in[1 + 4 * l + 1];
        bih[l] = (const float*)d_in[1 + 4 * l + 2];
        bhh[l] = (const float*)d_in[1 + 4 * l + 3];
    }

    const size_t T = SEQ_T;
    float* ws  = (float*)d_ws;
    float* xg  = ws;                 // T*192 floats (layers 0-3 use stride 48)
    float* yA  = xg + T * 192;       // T*16
    float* yB  = yA + T * 16;        // T*16
    float* h10 = yB + T * 16;        // 16  (c1l0 final hidden)
    float* h11 = h10 + 16;           // 16  (c1l1 final hidden)

    const dim3 blk(128);
    const int mBlocks = (int)(T / 16 / 4);  // 1024

    // compress1 layer 0: K=300, N=48, h0=ones, save hT -> h10
    xg_gemm_wmma<<<dim3(mBlocks, 3), blk, 0, stream>>>(input, Wih[0], bih[0], xg, 300, 48);
    gru_scan16<<<1, 32, 0, stream>>>(xg, Whh[0], bhh[0], nullptr, yA, h10);

    // compress1 layer 1: K=16, h0=ones, save hT -> h11
    xg_gemm_wmma<<<dim3(mBlocks, 3), blk, 0, stream>>>(yA, Wih[1], bih[1], xg, 16, 48);
    gru_scan16<<<1, 32, 0, stream>>>(xg, Whh[1], bhh[1], nullptr, yB, h11);

    // decode1 layer 0: h0 = h10
    xg_gemm_wmma<<<dim3(mBlocks, 3), blk, 0, stream>>>(yB, Wih[2], bih[2], xg, 16, 48);
    gru_scan16<<<1, 32, 0, stream>>>(xg, Whh[2], bhh[2], h10, yA, nullptr);

    // decode1 layer 1: h0 = h11
    xg_gemm_wmma<<<dim3(mBlocks, 3), blk, 0, stream>>>(yA, Wih[3], bih[3], xg, 16, 48);
    gru_scan16<<<1, 32, 0, stream>>>(xg, Whh[3], bhh[3], h11, yB, nullptr);

    // decode2: H=64, N=192, h0=ones; final hidden -> d_out (64 f32)
    xg_gemm_wmma<<<dim3(mBlocks, 12), blk, 0, stream>>>(yB, Wih[4], bih[4], xg, 16, 192);
    gru_scan64<<<1, 192, 0, stream>>>(xg, Whh[4], bhh[4], (float*)d_out);
}